// helmholtz_9010841387190
// MI455X (gfx1250) — compile-verified
//
#include <hip/hip_runtime.h>

// ---------------------------------------------------------------------------
// Multigrid Helmholtz V-cycle for MI455X (gfx1250).
//  * Stencil (relax / rescal): bandwidth-bound f32; working set of the finest
//    level (134 MB) fits the 192 MB global L2 -> simple coalesced kernels run
//    at L2 bandwidth. No LDS staging needed.
//  * Prolongation einsum (Ay . coarse . Ax^T): the one dense matmul in the
//    workload -> V_WMMA_F32_16X16X4_F32, one wave per 16x16 output tile,
//    4 waves per block, branchless K-loop (nc % 4 == 1 for every level, so
//    only the final K-group needs masking; addresses are clamped so all loads
//    are unconditional and predication is pure v_cndmask).
// ---------------------------------------------------------------------------

typedef float v2f __attribute__((ext_vector_type(2)));
typedef float v8f __attribute__((ext_vector_type(8)));

#define MREFIN 6

__device__ __forceinline__ int imin(int a, int b) { return a < b ? a : b; }

// Row i, column j of interp_matrix(nf, nc) with nf == 2*nc-1:
// pos = i*0.5 ; i0 = clip(floor(pos), 0, nc-2) ; weights (1-f, f) at (i0, i0+1)
__device__ __forceinline__ float interpW(int i, int j, int nc) {
  int i0 = i >> 1;
  if (i0 > nc - 2) i0 = nc - 2;
  float f = 0.5f * (float)i - (float)i0;
  float w = 0.0f;
  w = (j == i0)     ? 1.0f - f : w;
  w = (j == i0 + 1) ? f        : w;
  return w;
}

// psi0 = PSIGUESS ; q0 = Q * h0^2
__global__ void mg_init_kernel(const float* __restrict__ psig,
                               const float* __restrict__ Q,
                               float* __restrict__ psi0,
                               float* __restrict__ q0,
                               float h0sq, size_t n) {
  size_t i = (size_t)blockIdx.x * blockDim.x + threadIdx.x;
  if (i < n) {
    psi0[i] = psig[i];
    q0[i]   = Q[i] * h0sq;
  }
}

// Jacobi relax: dst interior = inv*(N+S+W+E - q), boundary = 0.
// srcZero=1 skips neighbor loads (first down-sweep relax of zeroed levels).
__global__ void relax_kernel(float* __restrict__ dst,
                             const float* __restrict__ src,
                             const float* __restrict__ q,
                             float inv, int n, int srcZero) {
  int x = blockIdx.x * blockDim.x + threadIdx.x;
  if (x >= n) return;
  int y = blockIdx.y;
  int b = blockIdx.z;
  size_t base = ((size_t)b * n + y) * n + x;
  float v = 0.0f;
  if (x > 0 && x < n - 1 && y > 0 && y < n - 1) {
    // cover first-touch HBM latency; L2-resident afterwards (gfx1250 prefetch)
    __builtin_prefetch(q + base + n, 0, 0);
    float s = 0.0f;
    if (!srcZero)
      s = src[base - n] + src[base + n] + src[base - 1] + src[base + 1];
    v = inv * (s - q[base]);
  }
  dst[base] = v;
}

// Restriction: qc interior = 4*(c*psi[2y][2x] - up - dn - lf - rt + q[2y][2x])
__global__ void rescal_kernel(float* __restrict__ qc,
                              const float* __restrict__ psi,
                              const float* __restrict__ q,
                              float c, int nf, int ncg) {
  int x = blockIdx.x * blockDim.x + threadIdx.x;
  if (x >= ncg) return;
  int y = blockIdx.y;
  int b = blockIdx.z;
  size_t ob = ((size_t)b * ncg + y) * ncg + x;
  float v = 0.0f;
  if (x > 0 && x < ncg - 1 && y > 0 && y < ncg - 1) {
    size_t fb = ((size_t)b * nf + 2 * y) * nf + 2 * x;
    v = 4.0f * (c * psi[fb] - psi[fb - nf] - psi[fb + nf]
                - psi[fb - 1] - psi[fb + 1] + q[fb]);
  }
  qc[ob] = v;
}

// ---------------------------------------------------------------------------
// WMMA f32 16x16x4 fragment layout (ISA 7.12.2):
//   A (16x4): lanes 0-15 carry K=0,1 in v[0],v[1]; lanes 16-31 carry K=2,3.
//   B (4x16): mirrored (N across lanes, same K split).
//   C/D     : VGPR r <-> rows r (lanes 0-15) / r+8 (lanes 16-31), N = lane&15.
// ---------------------------------------------------------------------------

// Prolong stage 1: T[b, j, l] = sum_k coarse[b, j, k] * Ax[l, k]
__global__ void prolong_stage1(const float* __restrict__ coarse,
                               float* __restrict__ T,
                               int nc, int nf) {
  int lane = threadIdx.x & 31;
  int wave = threadIdx.x >> 5;
  int half = lane >> 4;
  int l16  = lane & 15;
  int b  = blockIdx.z;
  int mt = blockIdx.y * 4 + wave;        // tile over coarse rows (nc)
  int nt = blockIdx.x;                   // tile over output cols (nf)
  if (mt * 16 >= nc) return;             // wave-uniform exit (before any WMMA)

  int m   = mt * 16 + l16;
  int col = nt * 16 + l16;
  float aMask = (m < nc)   ? 1.0f : 0.0f;
  float bMask = (col < nf) ? 1.0f : 0.0f;
  const float* crow = coarse + ((size_t)b * nc + imin(m, nc - 1)) * nc;

  v8f acc = {};
  int ncMain = nc & ~3;                  // every level has nc % 4 == 1
  for (int k0 = 0; k0 < ncMain; k0 += 4) {
    int ka = k0 + 2 * half;              // ka, ka+1 < nc guaranteed
    v2f a, bb;
    a.x  = crow[ka]     * aMask;
    a.y  = crow[ka + 1] * aMask;
    bb.x = interpW(col, ka,     nc) * bMask;
    bb.y = interpW(col, ka + 1, nc) * bMask;
    acc = __builtin_amdgcn_wmma_f32_16x16x4_f32(false, a, false, bb,
                                                (short)0, acc, false, false);
  }
  for (int k0 = ncMain; k0 < nc; k0 += 4) {   // tail: one K-group, masked
    int ka = k0 + 2 * half;
    float v0 = crow[imin(ka,     nc - 1)];    // clamped addr: always valid
    float v1 = crow[imin(ka + 1, nc - 1)];
    v2f a, bb;
    a.x  = (ka     < nc) ? v0 * aMask : 0.0f;
    a.y  = (ka + 1 < nc) ? v1 * aMask : 0.0f;
    bb.x = (ka     < nc) ? interpW(col, ka,     nc) * bMask : 0.0f;
    bb.y = (ka + 1 < nc) ? interpW(col, ka + 1, nc) * bMask : 0.0f;
    acc = __builtin_amdgcn_wmma_f32_16x16x4_f32(false, a, false, bb,
                                                (short)0, acc, false, false);
  }
  if (col < nf) {
#pragma unroll
    for (int r = 0; r < 8; ++r) {
      int row = mt * 16 + r + 8 * half;
      if (row < nc) T[((size_t)b * nc + row) * nf + col] = acc[r];
    }
  }
}

// Prolong stage 2: fine[b, i, l] += sum_j Ay[i, j] * T[b, j, l]
// Accumulation folded into the WMMA C operand (existing fine tile preloaded).
__global__ void prolong_stage2(const float* __restrict__ T,
                               float* __restrict__ fine,
                               int nc, int nf) {
  int lane = threadIdx.x & 31;
  int wave = threadIdx.x >> 5;
  int half = lane >> 4;
  int l16  = lane & 15;
  int b  = blockIdx.z;
  int mt = blockIdx.y * 4 + wave;        // tile over fine rows (nf)
  int nt = blockIdx.x;                   // tile over fine cols (nf)
  if (mt * 16 >= nf) return;             // wave-uniform exit

  int m   = mt * 16 + l16;
  int col = nt * 16 + l16;
  float aMask = (m < nf)   ? 1.0f : 0.0f;
  float bMask = (col < nf) ? 1.0f : 0.0f;
  int colC = imin(col, nf - 1);
  const float* Tcol    = T    + (size_t)b * nc * nf + colC;
  const float* fineCol = fine + (size_t)b * nf * nf + colC;

  v8f acc = {};
#pragma unroll
  for (int r = 0; r < 8; ++r) {          // C operand: clamped-address preload
    int row = mt * 16 + r + 8 * half;
    float v = fineCol[(size_t)imin(row, nf - 1) * nf];
    acc[r] = (row < nf && col < nf) ? v : 0.0f;
  }

  int ncMain = nc & ~3;
  for (int k0 = 0; k0 < ncMain; k0 += 4) {
    int ka = k0 + 2 * half;
    v2f a, bb;
    a.x  = interpW(m, ka,     nc) * aMask;
    a.y  = interpW(m, ka + 1, nc) * aMask;
    bb.x = Tcol[(size_t)ka * nf]       * bMask;
    bb.y = Tcol[(size_t)(ka + 1) * nf] * bMask;
    acc = __builtin_amdgcn_wmma_f32_16x16x4_f32(false, a, false, bb,
                                                (short)0, acc, false, false);
  }
  for (int k0 = ncMain; k0 < nc; k0 += 4) {   // masked tail, clamped addrs
    int ka = k0 + 2 * half;
    float t0 = Tcol[(size_t)imin(ka,     nc - 1) * nf];
    float t1 = Tcol[(size_t)imin(ka + 1, nc - 1) * nf];
    v2f a, bb;
    a.x  = (ka     < nc) ? interpW(m, ka,     nc) * aMask : 0.0f;
    a.y  = (ka + 1 < nc) ? interpW(m, ka + 1, nc) * aMask : 0.0f;
    bb.x = (ka     < nc) ? t0 * bMask : 0.0f;
    bb.y = (ka + 1 < nc) ? t1 * bMask : 0.0f;
    acc = __builtin_amdgcn_wmma_f32_16x16x4_f32(false, a, false, bb,
                                                (short)0, acc, false, false);
  }
  if (col < nf) {
#pragma unroll
    for (int r = 0; r < 8; ++r) {
      int row = mt * 16 + r + 8 * half;
      if (row < nf) fine[((size_t)b * nf + row) * nf + col] = acc[r];
    }
  }
}

extern "C" void kernel_launch(void* const* d_in, const int* in_sizes, int n_in,
                              void* d_out, int out_size, void* d_ws, size_t ws_size,
                              hipStream_t stream) {
  const float* PSIG = (const float*)d_in[0];
  const float* Q    = (const float*)d_in[1];
  float* psi0 = (float*)d_out;                 // psis[0] lives in d_out

  const int B = 64;
  const int S[MREFIN] = {513, 257, 129, 65, 33, 17};

  double Hh[MREFIN];
  float invv[MREFIN], cc[MREFIN];
  for (int k = 0; k < MREFIN; ++k) {
    Hh[k]   = 0.5 / (double)(1 << (MREFIN - k - 1));      // H1 / 2^(MREFIN-k-1)
    invv[k] = (float)(1.0 / (4.0 + Hh[k] * Hh[k]));
    cc[k]   = (float)(4.0 + Hh[k] * Hh[k]);
  }

  // Workspace carve-up (floats): tmp (level-0 sized ping-pong, reused as the
  // GEMM intermediate T), q[0..5], psi[1..5].  Total ~172 MB.
  float* ws  = (float*)d_ws;
  size_t off = 0;
  float* tmp = ws;  off += (size_t)B * S[0] * S[0];
  float* qb[MREFIN];
  float* psiA[MREFIN];
  psiA[0] = psi0;
  for (int k = 0; k < MREFIN; ++k) { qb[k]   = ws + off; off += (size_t)B * S[k] * S[k]; }
  for (int k = 1; k < MREFIN; ++k) { psiA[k] = ws + off; off += (size_t)B * S[k] * S[k]; }
  float* Tbuf = tmp;   // lifetimes of relax ping-pong and GEMM T never overlap

  {
    size_t n = (size_t)B * S[0] * S[0];
    float h0sq = (float)(Hh[0] * Hh[0]);
    mg_init_kernel<<<dim3((unsigned)((n + 255) / 256)), dim3(256), 0, stream>>>(
        PSIG, Q, psi0, qb[0], h0sq, n);
  }

  const int NCYC = 2;
  for (int cyc = 0; cyc < NCYC; ++cyc) {
    // ---- down sweep -----------------------------------------------------
    for (int k = 0; k < MREFIN; ++k) {
      int n = S[k];
      dim3 blk(128, 1, 1);
      dim3 grd((n + 127) / 128, n, B);
      // psis[k] zeroed each cycle for k>0 -> first relax treats src as 0
      relax_kernel<<<grd, blk, 0, stream>>>(tmp, psiA[k], qb[k], invv[k], n, (k > 0) ? 1 : 0);
      relax_kernel<<<grd, blk, 0, stream>>>(psiA[k], tmp, qb[k], invv[k], n, 0);
      if (k < MREFIN - 1) {
        int ng = S[k + 1];
        dim3 g2((ng + 127) / 128, ng, B);
        rescal_kernel<<<g2, blk, 0, stream>>>(qb[k + 1], psiA[k], qb[k], cc[k], n, ng);
      }
    }
    // ---- up sweep -------------------------------------------------------
    for (int k = 0; k < MREFIN; ++k) {
      int kk = MREFIN - 1 - k;
      int n = S[kk];
      dim3 blk(128, 1, 1);
      dim3 grd((n + 127) / 128, n, B);
      relax_kernel<<<grd, blk, 0, stream>>>(tmp, psiA[kk], qb[kk], invv[kk], n, 0);
      relax_kernel<<<grd, blk, 0, stream>>>(psiA[kk], tmp, qb[kk], invv[kk], n, 0);
      if (k > 0) {
        int nc = S[k], nf = S[k - 1];
        int tilesM1 = (nc + 15) / 16;
        int tilesM2 = (nf + 15) / 16;
        dim3 g1((nf + 15) / 16, (tilesM1 + 3) / 4, B);
        prolong_stage1<<<g1, dim3(128), 0, stream>>>(psiA[k], Tbuf, nc, nf);
        dim3 g2((nf + 15) / 16, (tilesM2 + 3) / 4, B);
        prolong_stage2<<<g2, dim3(128), 0, stream>>>(Tbuf, psiA[k - 1], nc, nf);
      }
    }
  }
}